// SSIMLoss_62062277427213
// MI455X (gfx1250) — compile-verified
//
#include <hip/hip_runtime.h>

// ---------------------------------------------------------------------------
// SSIM loss, MI455X (gfx1250). Memory-bound (~100MB in, 64B out, 23.3TB/s
// floor ~4.3us). Single-pass tiled kernel; vertical 11-tap box filter done on
// the matrix unit via V_WMMA_F32_16X16X4_F32 with a banded 0/1 A-matrix.
// ---------------------------------------------------------------------------

typedef float v2f __attribute__((ext_vector_type(2)));
typedef float v8f __attribute__((ext_vector_type(8)));

#define WIN   11
#define IMG_H 512
#define IMG_W 512
#define OUT_H 502
#define OUT_W 502
#define NB    16
#define NC    3

#define SSIM_C1 6.5025f    // (0.01*255)^2
#define SSIM_C2 58.5225f   // (0.03*255)^2

// Tile config: one workgroup = 16 output rows x 128 output cols of one plane.
#define TR       16
#define TC       128
#define IN_R     (TR + WIN - 1)   // 26 staged input rows
#define IN_CPAD  140              // >= TC+10=138, padded
#define S_ROWS   28               // 26 + 2 zero rows (K padded to 7 chunks of 4)
#define S_STRIDE 132              // 128 cols + pad

#define TILES_R  32               // ceil(502/16)
#define TILES_C  4                // ceil(502/128)
#define PART_PER_BATCH (NC * TILES_R * TILES_C)   // 384

__global__ __launch_bounds__(256)
void ssim_tile_kernel(const float* __restrict__ img1,
                      const float* __restrict__ img2,
                      float* __restrict__ partial)
{
    __shared__ float rawX[IN_R][IN_CPAD];
    __shared__ float rawY[IN_R][IN_CPAD];
    __shared__ float S[5][S_ROWS][S_STRIDE];   // row-filtered streams
    __shared__ float red[256];

    const int tc    = blockIdx.x;          // 0..3
    const int tr    = blockIdx.y;          // 0..31
    const int plane = blockIdx.z;          // b*3 + c, 0..47
    const int tid   = threadIdx.x;

    const int r0 = tr * TR;                // output-row base
    const int c0 = tc * TC;                // output-col base
    const size_t base = (size_t)plane * IMG_H * IMG_W;

    // ---- Stage A: stage raw tiles (edge-clamped; invalid outputs masked later)
    for (int i = tid; i < IN_R * IN_CPAD; i += 256) {
        int rr = i / IN_CPAD, cc = i % IN_CPAD;
        int gr = min(r0 + rr, IMG_H - 1);
        int gc = min(c0 + cc, IMG_W - 1);
        size_t idx = base + (size_t)gr * IMG_W + gc;
        rawX[rr][cc] = img1[idx];
        rawY[rr][cc] = img2[idx];
    }
    __syncthreads();

    // ---- Stage B: horizontal 11-tap sliding sums for the 5 streams.
    // Work unit = (row 0..25, stream 0..4, half 0..1), each covers 64 cols.
    for (int u = tid; u < IN_R * 5 * 2; u += 256) {
        const int h = u & 1;
        const int s = (u >> 1) % 5;
        const int r = u / 10;
        const int cb = h * 64;
        auto pval = [&](int c) -> float {
            float x = rawX[r][c], y = rawY[r][c];
            switch (s) {
                case 0:  return x;
                case 1:  return y;
                case 2:  return x * x;
                case 3:  return y * y;
                default: return x * y;
            }
        };
        float sum = 0.f;
        #pragma unroll
        for (int d = 0; d < WIN; ++d) sum += pval(cb + d);
        S[s][r][cb] = sum;
        for (int j = 1; j < 64; ++j) {
            sum += pval(cb + j + WIN - 1) - pval(cb + j - 1);
            S[s][r][cb + j] = sum;
        }
    }
    // Zero the two K-padding rows (A band is also 0 there; keep B finite).
    for (int i = tid; i < 5 * 2 * S_STRIDE; i += 256) {
        int s   = i / (2 * S_STRIDE);
        int rem = i % (2 * S_STRIDE);
        S[s][IN_R + rem / S_STRIDE][rem % S_STRIDE] = 0.f;
    }
    __syncthreads();

    // ---- Stage C: vertical 11-tap sum as banded matmul on the matrix unit.
    // Out(16x16) = Band(16x26) * S(26x16), K padded to 28, 7 chunks of 4.
    // f32 WMMA layouts (wave32):
    //   A 16x4 : lanes 0-15 -> M=lane, K={k0,k0+1}; lanes16-31 -> K={k0+2,k0+3}
    //   B 4x16 : lanes 0-15 -> N=lane, K rows {k0,k0+1}; lanes16-31 -> {k0+2,k0+3}
    //   D 16x16: vgpr j, lane L -> M = j + (L>=16 ? 8:0), N = L%16
    const int lane = tid & 31;
    const int wave = tid >> 5;
    const int n0   = wave * 16;          // column subtile base (8 waves * 16 = 128)
    const int l15  = lane & 15;
    const int hi   = lane >> 4;          // 0 or 1
    const int ka_off = hi * 2;

    v8f zero8 = {0.f, 0.f, 0.f, 0.f, 0.f, 0.f, 0.f, 0.f};
    v8f acc[5];
    #pragma unroll
    for (int s = 0; s < 5; ++s) acc[s] = zero8;

    #pragma unroll
    for (int ch = 0; ch < 7; ++ch) {
        const int ka = ch * 4 + ka_off;
        v2f a;
        a.x = (ka     >= l15 && ka     <= l15 + 10 && ka     < IN_R) ? 1.f : 0.f;
        a.y = (ka + 1 >= l15 && ka + 1 <= l15 + 10 && ka + 1 < IN_R) ? 1.f : 0.f;
        #pragma unroll
        for (int s = 0; s < 5; ++s) {
            v2f b;
            b.x = S[s][ka][n0 + l15];
            b.y = S[s][ka + 1][n0 + l15];
            acc[s] = __builtin_amdgcn_wmma_f32_16x16x4_f32(
                false, a, false, b, (short)0, acc[s], false, false);
        }
    }

    // ---- Stage D: fused SSIM pointwise math (all 5 tiles share the D layout).
    const float inv = 1.0f / (float)(WIN * WIN);
    float lsum = 0.f;
    #pragma unroll
    for (int j = 0; j < 8; ++j) {
        const int M    = j + hi * 8;
        const int outr = r0 + M;
        const int outc = c0 + n0 + l15;
        float mu1 = acc[0][j] * inv;
        float mu2 = acc[1][j] * inv;
        float ex2 = acc[2][j] * inv;
        float ey2 = acc[3][j] * inv;
        float exy = acc[4][j] * inv;
        float mu1s = mu1 * mu1, mu2s = mu2 * mu2, m12 = mu1 * mu2;
        float s1  = ex2 - mu1s;
        float s2  = ey2 - mu2s;
        float s12 = exy - m12;
        float nume = (2.f * m12 + SSIM_C1) * (2.f * s12 + SSIM_C2);
        float deno = (mu1s + mu2s + SSIM_C1) * (s1 + s2 + SSIM_C2);
        float v = nume / deno;
        if (outr < OUT_H && outc < OUT_W) lsum += v;
    }

    // ---- Deterministic block reduction -> one partial per workgroup.
    red[tid] = lsum;
    __syncthreads();
    #pragma unroll
    for (int s = 128; s > 0; s >>= 1) {
        if (tid < s) red[tid] += red[tid + s];
        __syncthreads();
    }
    if (tid == 0) {
        int pidx = (plane * TILES_R + tr) * TILES_C + tc;
        partial[pidx] = red[0];
    }
}

__global__ __launch_bounds__(256)
void ssim_finalize_kernel(const float* __restrict__ partial,
                          float* __restrict__ out)
{
    __shared__ float red[256];
    const int b   = blockIdx.x;     // 0..15
    const int tid = threadIdx.x;
    float s = 0.f;
    for (int i = tid; i < PART_PER_BATCH; i += 256)
        s += partial[b * PART_PER_BATCH + i];
    red[tid] = s;
    __syncthreads();
    #pragma unroll
    for (int k = 128; k > 0; k >>= 1) {
        if (tid < k) red[tid] += red[tid + k];
        __syncthreads();
    }
    if (tid == 0)
        out[b] = 1.0f - red[0] / ((float)NC * OUT_H * OUT_W);
}

extern "C" void kernel_launch(void* const* d_in, const int* in_sizes, int n_in,
                              void* d_out, int out_size, void* d_ws, size_t ws_size,
                              hipStream_t stream) {
    const float* img1 = (const float*)d_in[0];
    const float* img2 = (const float*)d_in[1];
    float* out     = (float*)d_out;
    float* partial = (float*)d_ws;   // NB * PART_PER_BATCH = 6144 floats (24 KB)

    dim3 grid(TILES_C, TILES_R, NB * NC);
    ssim_tile_kernel<<<grid, 256, 0, stream>>>(img1, img2, partial);
    ssim_finalize_kernel<<<NB, 256, 0, stream>>>(partial, out);
}